// OmniAnomaly_2439541424422
// MI455X (gfx1250) — compile-verified
//
#include <hip/hip_runtime.h>
#include <math.h>

// ---------------------------------------------------------------------------
// OmniAnomaly VAE forward for MI455X (gfx1250, wave32).
//
// Roofline: ~69 MB mandatory HBM traffic (Wih0 50.3MB + Wd3 16.8MB + x/bd3/xr)
// vs ~34 MFLOP -> purely bandwidth bound (~3us @ 23.3 TB/s). Keep fp32, spread
// the two big matvecs across the machine, do the tiny middle serially.
// Matrix ops use V_WMMA_F32_16X16X4_F32 (full-precision CDNA5 WMMA).
//
// Both GRU cells see h == 0 -> Whh0/Whh1 never read (saves 2 weight reads).
// ---------------------------------------------------------------------------

typedef float v2f __attribute__((ext_vector_type(2)));
typedef float v4f __attribute__((ext_vector_type(4)));
typedef float v8f __attribute__((ext_vector_type(8)));

#define FEATS     131072
#define NH        32
#define NL        8
#define NCHUNK    256                   // K-chunks for Wih0 @ x
#define CHUNK_LEN (FEATS / NCHUNK)      // 512 floats per chunk
#define ROW_TILES 6                     // 96 rows / 16

__device__ __forceinline__ float sigmoidf_(float t) {
    return 1.0f / (1.0f + expf(-t));
}

// One wave computes 16 rows x CHUNK_LEN partial dot products of W @ x using
// V_WMMA_F32_16X16X4_F32.
//   A (16x4 fp32, documented layout): lanes 0-15 hold K0(v0),K1(v1);
//                                     lanes 16-31 hold K2(v0),K3(v1)
//     -> per-lane b64 load at  wrow[k + 2*(lane>>4)]
//   B (4x16): column 0 = x[k..k+3], columns 1..15 = 0 (they cannot affect
//     C column 0). Row k of B striped across lanes within a VGPR:
//     v0: lane0=K0, lane16=K1 ; v1: lane0=K2, lane16=K3.
//   C (16x16): column 0 read back from lane 0 (M=0..7) and lane 16 (M=8..15).
__global__ void gi0_partial_kernel(const float* __restrict__ W,
                                   const float* __restrict__ x,
                                   float* __restrict__ partial) {
    // Force scalar wave id -> tile/chunk/kbase are SGPRs, loop becomes a
    // uniform s_cbranch loop and x addresses become wave-uniform.
    const int wave = __builtin_amdgcn_readfirstlane(
        blockIdx.x * (blockDim.x >> 5) + (threadIdx.x >> 5));
    const int lane   = threadIdx.x & 31;
    const int tile   = wave % ROW_TILES;       // which 16-row block of Wih0
    const int chunk  = wave / ROW_TILES;       // which K segment
    const int khalf  = lane >> 4;              // 0: K0/K1  1: K2/K3
    const int lolane = ((lane & 15) == 0);     // lanes 0 and 16 carry B col 0

    const float* wrow  = W + (size_t)(tile * 16 + (lane & 15)) * FEATS;
    const int    kbase = chunk * CHUNK_LEN;

    v8f c = {};
#pragma unroll 8
    for (int k = kbase; k < kbase + CHUNK_LEN; k += 4) {
        v2f a = *(const v2f*)(wrow + k + 2 * khalf);   // global_load_b64
        v4f xq = *(const v4f*)(x + k);                 // uniform 16B load
        float bx = khalf ? xq.y : xq.x;    // lane0 -> x[k],   lane16 -> x[k+1]
        float by = khalf ? xq.w : xq.z;    // lane0 -> x[k+2], lane16 -> x[k+3]
        v2f b;
        b.x = lolane ? bx : 0.0f;
        b.y = lolane ? by : 0.0f;
        c = __builtin_amdgcn_wmma_f32_16x16x4_f32(
                false, a, false, b, (short)0, c, false, false);
    }

    if (lolane) {
        // C column 0: lane 0 holds M=0..7 in c[0..7], lane 16 holds M=8..15.
        // partial layout: [chunk][row] -> two contiguous b128 stores per lane.
        float* dst = partial + (size_t)chunk * 96 + tile * 16 + khalf * 8;
        v4f lo = {c[0], c[1], c[2], c[3]};
        v4f hi = {c[4], c[5], c[6], c[7]};
        *(v4f*)(dst)     = lo;
        *(v4f*)(dst + 4) = hi;
    }
}

// Single-block serial middle: reduce partials -> GRU0 -> GRU1 -> encoder MLP
// -> reparameterize -> decoder MLP (up to d2).
__global__ void middle_kernel(const float* __restrict__ partial,
                              const float* __restrict__ bih0, const float* __restrict__ bhh0,
                              const float* __restrict__ Wih1, const float* __restrict__ bih1,
                              const float* __restrict__ bhh1,
                              const float* __restrict__ We1,  const float* __restrict__ be1,
                              const float* __restrict__ ae1,
                              const float* __restrict__ We2,  const float* __restrict__ be2,
                              const float* __restrict__ ae2,
                              const float* __restrict__ We3,  const float* __restrict__ be3,
                              const float* __restrict__ Wd1,  const float* __restrict__ bd1,
                              const float* __restrict__ ad1,
                              const float* __restrict__ Wd2,  const float* __restrict__ bd2,
                              const float* __restrict__ ad2,
                              const float* __restrict__ eps,
                              float* __restrict__ out_tail,   // d_out + FEATS
                              float* __restrict__ dvec) {     // ws: 32 floats
    __shared__ float g[96], h1[NH], h2[NH], e1s[NH], e2s[NH], st[2 * NL],
                     zl[NL], d1s[NH];
    const int t = threadIdx.x;

    if (t < 96) {                                   // gi0 = Wih0@x + bih0
        float acc = bih0[t];
        for (int c2 = 0; c2 < NCHUNK; ++c2) acc += partial[(size_t)c2 * 96 + t];
        g[t] = acc;
    }
    __syncthreads();
    if (t < NH) {                                   // GRU cell 0 (h = 0)
        float r = sigmoidf_(g[t]      + bhh0[t]);
        float z = sigmoidf_(g[32 + t] + bhh0[32 + t]);
        float n = tanhf(g[64 + t] + r * bhh0[64 + t]);
        h1[t] = (1.0f - z) * n;
    }
    __syncthreads();
    if (t < 96) {                                   // gi1 = Wih1@h1 + bih1
        float acc = bih1[t];
        for (int j = 0; j < NH; ++j) acc += Wih1[t * NH + j] * h1[j];
        g[t] = acc;
    }
    __syncthreads();
    if (t < NH) {                                   // GRU cell 1 (h = 0)
        float r = sigmoidf_(g[t]      + bhh1[t]);
        float z = sigmoidf_(g[32 + t] + bhh1[32 + t]);
        float n = tanhf(g[64 + t] + r * bhh1[64 + t]);
        h2[t] = (1.0f - z) * n;
    }
    __syncthreads();
    if (t < NH) {                                   // e = prelu(We1@h2 + be1)
        float acc = be1[t];
        for (int j = 0; j < NH; ++j) acc += We1[t * NH + j] * h2[j];
        e1s[t] = acc >= 0.0f ? acc : ae1[0] * acc;
    }
    __syncthreads();
    if (t < NH) {                                   // e = prelu(We2@e + be2)
        float acc = be2[t];
        for (int j = 0; j < NH; ++j) acc += We2[t * NH + j] * e1s[j];
        e2s[t] = acc >= 0.0f ? acc : ae2[0] * acc;
    }
    __syncthreads();
    if (t < 2 * NL) {                               // stats = We3@e + be3
        float acc = be3[t];
        for (int j = 0; j < NH; ++j) acc += We3[t * NH + j] * e2s[j];
        st[t] = acc;
    }
    __syncthreads();
    if (t < NL) {                                   // reparameterize
        float mu = st[t], lv = st[NL + t];
        out_tail[t]      = mu;                      // mu     @ FEATS + 0
        out_tail[NL + t] = lv;                      // logvar @ FEATS + 8
        zl[t] = mu + eps[t] * expf(0.5f * lv);
    }
    __syncthreads();
    if (t < NH) {                                   // d = prelu(Wd1@z + bd1)
        float acc = bd1[t];
        for (int j = 0; j < NL; ++j) acc += Wd1[t * NL + j] * zl[j];
        d1s[t] = acc >= 0.0f ? acc : ad1[0] * acc;
    }
    __syncthreads();
    if (t < NH) {                                   // d = prelu(Wd2@d + bd2)
        float acc = bd2[t];
        for (int j = 0; j < NH; ++j) acc += Wd2[t * NH + j] * d1s[j];
        dvec[t] = acc >= 0.0f ? acc : ad2[0] * acc;
        out_tail[16 + t]      = h1[t];              // hidden[0] @ FEATS + 16
        out_tail[16 + NH + t] = h2[t];              // hidden[1] @ FEATS + 48
    }
}

// xr = sigmoid(Wd3 @ d + bd3): one wave per 16 output rows, 8 WMMA steps
// over K = 32. Same A/B/C mapping as gi0_partial_kernel.
__global__ void decoder_out_kernel(const float* __restrict__ Wd3,
                                   const float* __restrict__ bd3,
                                   const float* __restrict__ dvec,
                                   float* __restrict__ xr) {
    const int tile = __builtin_amdgcn_readfirstlane(
        blockIdx.x * (blockDim.x >> 5) + (threadIdx.x >> 5));
    const int lane   = threadIdx.x & 31;
    const int khalf  = lane >> 4;
    const int lolane = ((lane & 15) == 0);

    const float* wrow = Wd3 + (size_t)(tile * 16 + (lane & 15)) * NH;

    v8f c = {};
#pragma unroll
    for (int k = 0; k < NH; k += 4) {
        v2f a  = *(const v2f*)(wrow + k + 2 * khalf);
        v4f xq = *(const v4f*)(dvec + k);              // uniform 16B load
        float bx = khalf ? xq.y : xq.x;
        float by = khalf ? xq.w : xq.z;
        v2f b;
        b.x = lolane ? bx : 0.0f;
        b.y = lolane ? by : 0.0f;
        c = __builtin_amdgcn_wmma_f32_16x16x4_f32(
                false, a, false, b, (short)0, c, false, false);
    }

    if (lolane) {
        const int r0 = tile * 16 + khalf * 8;
        v4f blo = *(const v4f*)(bd3 + r0);
        v4f bhi = *(const v4f*)(bd3 + r0 + 4);
        v4f olo = {sigmoidf_(c[0] + blo.x), sigmoidf_(c[1] + blo.y),
                   sigmoidf_(c[2] + blo.z), sigmoidf_(c[3] + blo.w)};
        v4f ohi = {sigmoidf_(c[4] + bhi.x), sigmoidf_(c[5] + bhi.y),
                   sigmoidf_(c[6] + bhi.z), sigmoidf_(c[7] + bhi.w)};
        *(v4f*)(xr + r0)     = olo;
        *(v4f*)(xr + r0 + 4) = ohi;
    }
}

extern "C" void kernel_launch(void* const* d_in, const int* in_sizes, int n_in,
                              void* d_out, int out_size, void* d_ws, size_t ws_size,
                              hipStream_t stream) {
    const float* x    = (const float*)d_in[0];
    const float* Wih0 = (const float*)d_in[1];
    // d_in[2] = Whh0 (unused: h == 0)
    const float* bih0 = (const float*)d_in[3];
    const float* bhh0 = (const float*)d_in[4];
    const float* Wih1 = (const float*)d_in[5];
    // d_in[6] = Whh1 (unused: h == 0)
    const float* bih1 = (const float*)d_in[7];
    const float* bhh1 = (const float*)d_in[8];
    const float* We1  = (const float*)d_in[9];
    const float* be1  = (const float*)d_in[10];
    const float* ae1  = (const float*)d_in[11];
    const float* We2  = (const float*)d_in[12];
    const float* be2  = (const float*)d_in[13];
    const float* ae2  = (const float*)d_in[14];
    const float* We3  = (const float*)d_in[15];
    const float* be3  = (const float*)d_in[16];
    const float* Wd1  = (const float*)d_in[17];
    const float* bd1  = (const float*)d_in[18];
    const float* ad1  = (const float*)d_in[19];
    const float* Wd2  = (const float*)d_in[20];
    const float* bd2  = (const float*)d_in[21];
    const float* ad2  = (const float*)d_in[22];
    const float* Wd3  = (const float*)d_in[23];
    const float* bd3  = (const float*)d_in[24];
    const float* eps  = (const float*)d_in[25];

    float* out     = (float*)d_out;
    float* ws      = (float*)d_ws;
    float* partial = ws;                 // NCHUNK * 96 floats (96 KB)
    float* dvec    = ws + NCHUNK * 96;   // 32 floats

    // 1) Wih0 @ x partials: 6 row-tiles x 256 K-chunks = 1536 waves.
    gi0_partial_kernel<<<(ROW_TILES * NCHUNK) / 8, 256, 0, stream>>>(
        Wih0, x, partial);

    // 2) Serial middle network (reduction + GRUs + VAE up to decoder d2).
    middle_kernel<<<1, 128, 0, stream>>>(
        partial, bih0, bhh0, Wih1, bih1, bhh1,
        We1, be1, ae1, We2, be2, ae2, We3, be3,
        Wd1, bd1, ad1, Wd2, bd2, ad2, eps,
        out + FEATS, dvec);

    // 3) xr = sigmoid(Wd3 @ d + bd3): 8192 row-tiles, 8 waves per block.
    decoder_out_kernel<<<(FEATS / 16) / 8, 256, 0, stream>>>(
        Wd3, bd3, dvec, out);
}